// LiteChunk1_44890998177974
// MI455X (gfx1250) — compile-verified
//
#include <hip/hip_runtime.h>
#include <math.h>

// ---------------- model constants ----------------
static constexpr int H    = 1024;
static constexpr int NH   = 8;
static constexpr int DFF  = 4096;
static constexpr int PLI  = 256;
static constexpr int CTX  = 2048;
static constexpr int MAXHD= 512;
static constexpr int NLAY = 15;
static constexpr int RPB  = 64;   // rows per GEMV block (divides every row_active)

// ---------------- CDNA5 vector types ----------------
typedef __attribute__((ext_vector_type(2))) float v2f;
typedef __attribute__((ext_vector_type(8))) float v8f;

// ---------------- helpers ----------------
__device__ __forceinline__ float block_sum(float v, float* sm) {
  int t = threadIdx.x;
  sm[t] = v; __syncthreads();
  for (int s = blockDim.x >> 1; s > 0; s >>= 1) {
    if (t < s) sm[t] += sm[t + s];
    __syncthreads();
  }
  float r = sm[0]; __syncthreads();
  return r;
}

__device__ __forceinline__ float gelu_t(float x) {
  float u = 0.7978845608028654f * (x + 0.044715f * x * x * x);
  return 0.5f * x * (1.0f + tanhf(u));
}

// ---------------- kernels ----------------

// h = embed[tok]*sqrt(H)*(1-is_img) + img*is_img
__global__ void init_h_kernel(const float* __restrict__ embed,
                              const int* __restrict__ ids,
                              const float* __restrict__ img,
                              float* __restrict__ h) {
  __shared__ float sm[256];
  int t = threadIdx.x;
  float s = 0.f;
  for (int i = t; i < H; i += 256) s += fabsf(img[i]);
  s = block_sum(s, sm);
  float isimg = (s > 0.f) ? 1.f : 0.f;
  int tok = ids[0];
  const float* e = embed + (size_t)tok * H;
  for (int i = t; i < H; i += 256)
    h[i] = e[i] * 32.0f * (1.f - isimg) + img[i] * isimg;
}

// y = x * rsqrt(mean(x^2)+eps) * (1+w)
__global__ void rmsnorm_kernel(const float* __restrict__ x,
                               const float* __restrict__ w,
                               float* __restrict__ y, int n) {
  __shared__ float sm[256];
  int t = threadIdx.x;
  float s = 0.f;
  for (int i = t; i < n; i += 256) { float v = x[i]; s += v * v; }
  s = block_sum(s, sm);
  float sc = rsqrtf(s / (float)n + 1e-6f);
  for (int i = t; i < n; i += 256) y[i] = x[i] * sc * (1.f + w[i]);
}

// h = (h + rms(g,w)) * (ls ? ls[0] : 1)
__global__ void add_rms_kernel(const float* __restrict__ g,
                               const float* __restrict__ w,
                               float* __restrict__ h, int n,
                               const float* __restrict__ ls) {
  __shared__ float sm[256];
  int t = threadIdx.x;
  float s = 0.f;
  for (int i = t; i < n; i += 256) { float v = g[i]; s += v * v; }
  s = block_sum(s, sm);
  float sc = rsqrtf(s / (float)n + 1e-6f);
  float m = ls ? ls[0] : 1.f;
  for (int i = t; i < n; i += 256)
    h[i] = (h[i] + g[i] * sc * (1.f + w[i])) * m;
}

// GEMV partial: partial[rowgrp][j] = sum over RPB rows of x[i]*W[pr(i)][pc(j)]
// col mapping: physical col = (j/col_active)*col_group + j%col_active
// row mapping: physical row = (i/row_active)*row_group + i%row_active
// 32 threads, 4 consecutive cols/thread: pure b128 weight streaming with
// speculative global_prefetch_b8 running 16 rows ahead of the loads.
__global__ void gemv_part_kernel(const float* __restrict__ x,
                                 const float* __restrict__ W,
                                 float* __restrict__ partial,
                                 int cols_logical, int ldw,
                                 int col_active, int col_group,
                                 int row_active, int row_group) {
  const int t  = threadIdx.x;
  const int j0 = blockIdx.x * 128;
  const int r0 = blockIdx.y * RPB;
  const int pc0 = (j0 / col_active) * col_group + (j0 % col_active);
  const int pr0 = (r0 / row_active) * row_group + (r0 % row_active);
  const float4* Wp = reinterpret_cast<const float4*>(
      W + (size_t)pr0 * ldw + pc0) + t;
  const size_t ld4 = (size_t)(ldw >> 2);
  const float* xp = x + r0;
  float4 acc; acc.x = acc.y = acc.z = acc.w = 0.f;
#pragma unroll 8
  for (int i = 0; i < RPB; ++i) {
    if ((i & 7) == 0)
      __builtin_prefetch((const void*)(Wp + (size_t)(i + 16) * ld4), 0, 0);
    float xv = xp[i];
    float4 wv = Wp[(size_t)i * ld4];
    acc.x += xv * wv.x; acc.y += xv * wv.y;
    acc.z += xv * wv.z; acc.w += xv * wv.w;
  }
  float* out = partial + (size_t)blockIdx.y * cols_logical + j0 + t * 4;
  out[0] = acc.x; out[1] = acc.y; out[2] = acc.z; out[3] = acc.w;
}

__global__ void gemv_reduce_kernel(const float* __restrict__ partial,
                                   float* __restrict__ y,
                                   int cols, int ngroups) {
  int j = blockIdx.x * 256 + threadIdx.x;
  if (j < cols) {
    float s = 0.f;
    for (int g = 0; g < ngroups; ++g) s += partial[(size_t)g * cols + j];
    y[j] = s;
  }
}

// blocks 0..7: per-head q rms(qn_w)+rope -> qr (stride 512, rows 8..15 pre-zeroed)
// block 8: k rms(kn_w)+rope -> kr ; block 9: v vnorm in place
__global__ void norm_rope_kernel(const float* __restrict__ q,
                                 const float* __restrict__ kvec,
                                 float* __restrict__ vvec,
                                 float* __restrict__ qr,
                                 float* __restrict__ kr,
                                 const float* __restrict__ qnw,
                                 const float* __restrict__ knw,
                                 const float* __restrict__ cosT,
                                 const float* __restrict__ sinT,
                                 const int* __restrict__ posp, int hd) {
  __shared__ float sm[128];
  int t = threadIdx.x, b = blockIdx.x;
  int pos = posp[0];
  const float* cosr = cosT + (size_t)pos * hd;
  const float* sinr = sinT + (size_t)pos * hd;
  if (b < 9) {
    const float* src = (b < 8) ? (q + b * hd) : kvec;
    const float* w   = (b < 8) ? qnw : knw;
    float* dst       = (b < 8) ? (qr + b * MAXHD) : kr;
    float s = 0.f;
    for (int d = t; d < hd; d += 128) { float v = src[d]; s += v * v; }
    s = block_sum(s, sm);
    float sc = rsqrtf(s / (float)hd + 1e-6f);
    int half = hd >> 1;
    for (int d = t; d < hd; d += 128) {
      float xn = src[d] * sc * (1.f + w[d]);
      int dp = (d < half) ? d + half : d - half;
      float xp = src[dp] * sc * (1.f + w[dp]);
      float rot = (d < half) ? -xp : xp;
      dst[d] = xn * cosr[d] + rot * sinr[d];
    }
  } else {
    float s = 0.f;
    for (int d = t; d < hd; d += 128) { float v = vvec[d]; s += v * v; }
    s = block_sum(s, sm);
    float sc = rsqrtf(s / (float)hd + 1e-6f);
    for (int d = t; d < hd; d += 128) vvec[d] *= sc;
  }
}

// scores[m][n0..n0+15] for heads m=0..7 via V_WMMA_F32_16X16X4_F32.
// A tile = q rows (16 rows, heads 8..15 zero-padded), B tile = K^T columns.
// One wave per 16-position tile; EXEC all ones across all WMMA issues.
__global__ void attn_scores_wmma_kernel(const float* __restrict__ qr,
                                        const float* __restrict__ kcache,
                                        const float* __restrict__ kr,
                                        const float* __restrict__ mask,
                                        float* __restrict__ scores,
                                        const int* __restrict__ posp, int hd) {
  const int lane = threadIdx.x;          // 0..31
  const int n0   = blockIdx.x * 16;
  const int m    = lane & 15;
  const int half = lane >> 4;            // K sub-pair select
  const int pos  = posp[0];
  const int n    = n0 + m;
  // per-lane base row of K (row `pos` comes from the fresh roped k, not cache)
  const float* krow = (n == pos) ? kr : (kcache + (size_t)n * MAXHD);
  const float* arow = qr + (size_t)m * MAXHD;
  v8f c = {};
  for (int d0 = 0; d0 < hd; d0 += 4) {
    v2f a, b;
    // ISA 16x4 f32 A layout: lanes 0-15 hold K=0,1; lanes 16-31 hold K=2,3
    a.x = arow[d0 + 2 * half];
    a.y = arow[d0 + 2 * half + 1];
    // B (4x16) mirrored: lane holds column n, K=2*half / 2*half+1
    b.x = krow[d0 + 2 * half];
    b.y = krow[d0 + 2 * half + 1];
    c = __builtin_amdgcn_wmma_f32_16x16x4_f32(
        /*neg_a=*/false, a, /*neg_b=*/false, b,
        /*c_mod=*/(short)0, c, /*reuse_a=*/false, /*reuse_b=*/false);
  }
  float mv = mask[n0 + (lane & 15)];
  if (lane < 16) {
    // C/D layout: lanes 0-15 VGPR r holds M=r (heads 0..7), N=lane
#pragma unroll
    for (int r = 0; r < 8; ++r)
      scores[(size_t)r * CTX + n0 + lane] = c[r] + mv;
  }
}

__global__ void softmax_kernel(float* __restrict__ scores) {
  __shared__ float sm[256];
  int t = threadIdx.x;
  float* s = scores + (size_t)blockIdx.x * CTX;
  float mx = -1e30f;
  for (int n = t; n < CTX; n += 256) mx = fmaxf(mx, s[n]);
  sm[t] = mx; __syncthreads();
  for (int k = 128; k > 0; k >>= 1) {
    if (t < k) sm[t] = fmaxf(sm[t], sm[t + k]);
    __syncthreads();
  }
  mx = sm[0]; __syncthreads();
  float se = 0.f;
  for (int n = t; n < CTX; n += 256) {
    float e = __expf(s[n] - mx);
    s[n] = e; se += e;
  }
  se = block_sum(se, sm);
  float inv = 1.f / se;
  for (int n = t; n < CTX; n += 256) s[n] *= inv;
}

// attn[h][d] = sum_n p[n]*V[n][d], with row `pos` replaced by fresh v
__global__ void attn_v_kernel(const float* __restrict__ scores,
                              const float* __restrict__ vcache,
                              const float* __restrict__ vnew,
                              float* __restrict__ attn,
                              const int* __restrict__ posp, int hd) {
  int hh = blockIdx.y;
  int d = blockIdx.x * 128 + threadIdx.x;
  const float* p = scores + (size_t)hh * CTX;
  int pos = posp[0];
  float acc = 0.f;
#pragma unroll 8
  for (int n = 0; n < CTX; ++n)
    acc += p[n] * vcache[(size_t)n * MAXHD + d];
  acc += p[pos] * (vnew[d] - vcache[(size_t)pos * MAXHD + d]);
  attn[(size_t)hh * hd + d] = acc;
}

__global__ void gelu_mul_kernel(const float* __restrict__ a,
                                const float* __restrict__ b,
                                float* __restrict__ y, int n) {
  int i = blockIdx.x * 256 + threadIdx.x;
  if (i < n) y[i] = gelu_t(a[i]) * b[i];
}

// Emit K_new[:, :hd] / V_new[:, :hd] (cache with row pos replaced)
__global__ void kv_out_kernel(const float* __restrict__ kc,
                              const float* __restrict__ vc,
                              const float* __restrict__ kr,
                              const float* __restrict__ vnew,
                              const int* __restrict__ posp, int hd,
                              float* __restrict__ ok,
                              float* __restrict__ ov) {
  int n = blockIdx.x;
  int d = blockIdx.y * 128 + threadIdx.x;
  int pos = posp[0];
  ok[(size_t)n * hd + d] = (n == pos) ? kr[d] : kc[(size_t)n * MAXHD + d];
  ov[(size_t)n * hd + d] = (n == pos) ? vnew[d] : vc[(size_t)n * MAXHD + d];
}

__global__ void copy_kernel(const float* __restrict__ a,
                            float* __restrict__ b, int n) {
  int i = blockIdx.x * 256 + threadIdx.x;
  if (i < n) b[i] = a[i];
}

// ---------------- host-side GEMV helper ----------------
static void gemv(hipStream_t st, const float* x, const float* W,
                 float* partial, float* y,
                 int rows, int cols_logical, int ldw,
                 int col_active, int col_group,
                 int row_active, int row_group) {
  dim3 g(cols_logical / 128, rows / RPB);
  gemv_part_kernel<<<g, 32, 0, st>>>(x, W, partial, cols_logical, ldw,
                                     col_active, col_group, row_active, row_group);
  gemv_reduce_kernel<<<(cols_logical + 255) / 256, 256, 0, st>>>(
      partial, y, cols_logical, rows / RPB);
}

// ---------------- workspace layout (floats) ----------------
static constexpr size_t WS_H     = 0;        // 1024
static constexpr size_t WS_HS    = 1024;     // 1024
static constexpr size_t WS_Q     = 2048;     // 4096
static constexpr size_t WS_K     = 6144;     // 512
static constexpr size_t WS_V     = 6656;     // 512
static constexpr size_t WS_QR    = 7168;     // 16*512 = 8192 (zero-padded A tile)
static constexpr size_t WS_KR    = 15360;    // 512
static constexpr size_t WS_SC    = 15872;    // 8*2048 = 16384
static constexpr size_t WS_ATTN  = 32256;    // 4096
static constexpr size_t WS_O     = 36352;    // 1024
static constexpr size_t WS_GV    = 37376;    // 4096
static constexpr size_t WS_UV    = 41472;    // 4096
static constexpr size_t WS_GU    = 45568;    // 4096
static constexpr size_t WS_MLP   = 49664;    // 1024
static constexpr size_t WS_PV    = 50688;    // 256
static constexpr size_t WS_PG    = 50944;    // 256
static constexpr size_t WS_PLO   = 51200;    // 1024
static constexpr size_t WS_PART  = 52224;    // up to 65536 partial sums

extern "C" void kernel_launch(void* const* d_in, const int* in_sizes, int n_in,
                              void* d_out, int out_size, void* d_ws, size_t ws_size,
                              hipStream_t stream) {
  (void)in_sizes; (void)n_in; (void)out_size; (void)ws_size;
  const float* embed   = (const float*)d_in[0];
  const float* cos_s   = (const float*)d_in[1];
  const float* sin_s   = (const float*)d_in[2];
  const float* cos_f   = (const float*)d_in[3];
  const float* sin_f   = (const float*)d_in[4];
  const float* ln_in   = (const float*)d_in[5];
  const float* ln_pa   = (const float*)d_in[6];
  const float* ln_pff  = (const float*)d_in[7];
  const float* ln_pof  = (const float*)d_in[8];
  const float* ln_ppli = (const float*)d_in[9];
  const float* qn_w    = (const float*)d_in[10];
  const float* kn_w    = (const float*)d_in[11];
  const float* Wq      = (const float*)d_in[12];
  const float* Wk      = (const float*)d_in[13];
  const float* Wv      = (const float*)d_in[14];
  const float* Wo      = (const float*)d_in[15];
  const float* Wg      = (const float*)d_in[16];
  const float* Wu      = (const float*)d_in[17];
  const float* Wd      = (const float*)d_in[18];
  const float* Wpg     = (const float*)d_in[19];
  const float* Wpp     = (const float*)d_in[20];
  const float* lscal   = (const float*)d_in[21];
  const float* kvc     = (const float*)d_in[22];
  const float* plc     = (const float*)d_in[23];
  const float* img     = (const float*)d_in[24];
  const float* mask    = (const float*)d_in[25];
  const int*   ids     = (const int*)d_in[27];
  const int*   posp    = (const int*)d_in[28];
  float* out = (float*)d_out;
  float* ws  = (float*)d_ws;

  float* h    = ws + WS_H;
  float* hs   = ws + WS_HS;
  float* q    = ws + WS_Q;
  float* kvec = ws + WS_K;
  float* vvec = ws + WS_V;
  float* qr   = ws + WS_QR;
  float* kr   = ws + WS_KR;
  float* sc   = ws + WS_SC;
  float* attn = ws + WS_ATTN;
  float* o    = ws + WS_O;
  float* gv   = ws + WS_GV;
  float* uv   = ws + WS_UV;
  float* gu   = ws + WS_GU;
  float* mlpo = ws + WS_MLP;
  float* pv   = ws + WS_PV;
  float* pg   = ws + WS_PG;
  float* plo  = ws + WS_PLO;
  float* part = ws + WS_PART;

  init_h_kernel<<<1, 256, 0, stream>>>(embed, ids, img, h);

  for (int l = 0; l < NLAY; ++l) {
    const bool full = ((l + 1) % 5) == 0;
    const int hd = full ? 512 : 256;
    const float* cosT = full ? cos_f : cos_s;
    const float* sinT = full ? sin_f : sin_s;
    const float* kcl = kvc + (size_t)l * CTX * MAXHD;
    const float* vcl = kvc + (size_t)(NLAY + l) * CTX * MAXHD;

    // ---- attention block ----
    rmsnorm_kernel<<<1, 256, 0, stream>>>(h, ln_in + (size_t)l * H, hs, H);
    gemv(stream, hs, Wq + (size_t)l * H * NH * MAXHD, part, q,
         H, NH * hd, NH * MAXHD, hd, MAXHD, H, H);
    gemv(stream, hs, Wk + (size_t)l * H * MAXHD, part, kvec,
         H, hd, MAXHD, hd, MAXHD, H, H);
    gemv(stream, hs, Wv + (size_t)l * H * MAXHD, part, vvec,
         H, hd, MAXHD, hd, MAXHD, H, H);
    hipMemsetAsync(qr, 0, 16 * MAXHD * sizeof(float), stream);
    norm_rope_kernel<<<10, 128, 0, stream>>>(q, kvec, vvec, qr, kr,
        qn_w + (size_t)l * MAXHD, kn_w + (size_t)l * MAXHD,
        cosT, sinT, posp, hd);
    attn_scores_wmma_kernel<<<CTX / 16, 32, 0, stream>>>(
        qr, kcl, kr, mask, sc, posp, hd);
    softmax_kernel<<<NH, 256, 0, stream>>>(sc);
    attn_v_kernel<<<dim3(hd / 128, NH), 128, 0, stream>>>(
        sc, vcl, vvec, attn, posp, hd);
    if (l == 13) {
      kv_out_kernel<<<dim3(CTX, 256 / 128), 128, 0, stream>>>(
          kcl, vcl, kr, vvec, posp, 256, out + 1024, out + 525312);
    }
    if (l == 14) {
      kv_out_kernel<<<dim3(CTX, 512 / 128), 128, 0, stream>>>(
          kcl, vcl, kr, vvec, posp, 512, out + 1049600, out + 2098176);
    }
    gemv(stream, attn, Wo + (size_t)l * NH * MAXHD * H, part, o,
         NH * hd, H, H, H, H, hd, MAXHD);
    add_rms_kernel<<<1, 256, 0, stream>>>(o, ln_pa + (size_t)l * H, h, H, nullptr);

    // ---- FFN block ----
    rmsnorm_kernel<<<1, 256, 0, stream>>>(h, ln_pff + (size_t)l * H, hs, H);
    gemv(stream, hs, Wg + (size_t)l * H * DFF, part, gv, H, DFF, DFF, DFF, DFF, H, H);
    gemv(stream, hs, Wu + (size_t)l * H * DFF, part, uv, H, DFF, DFF, DFF, DFF, H, H);
    gelu_mul_kernel<<<DFF / 256, 256, 0, stream>>>(gv, uv, gu, DFF);
    gemv(stream, gu, Wd + (size_t)l * DFF * H, part, mlpo, DFF, H, H, H, H, DFF, DFF);
    add_rms_kernel<<<1, 256, 0, stream>>>(mlpo, ln_pof + (size_t)l * H, h, H, nullptr);

    // ---- per-layer-input block ----
    gemv(stream, h, Wpg + (size_t)l * H * PLI, part, pv, H, PLI, PLI, PLI, PLI, H, H);
    gelu_mul_kernel<<<1, 256, 0, stream>>>(pv, plc + (size_t)l * PLI, pg, PLI);
    gemv(stream, pg, Wpp + (size_t)l * PLI * H, part, plo, PLI, H, H, H, H, PLI, PLI);
    add_rms_kernel<<<1, 256, 0, stream>>>(plo, ln_ppli + (size_t)l * H, h, H,
                                          lscal + l);
  }

  copy_kernel<<<(H + 255) / 256, 256, 0, stream>>>(h, out, H);
}